// SoftPool_43533788513062
// MI455X (gfx1250) — compile-verified
//
#include <hip/hip_runtime.h>
#include <hip/hip_bf16.h>

// ---------------------------------------------------------------------------
// SoftPool-style region sorter for MI455X (gfx1250, wave32, WMMA).
// B=16, C=256, N=8192, REGIONS=16, CABINS=8, pnt = N/4 = 2048.
// ---------------------------------------------------------------------------

#define BB   16
#define CC   256
#define NN   8192
#define RR   16
#define PNT  2048      // NN / 4
#define CAB  8
#define PCAB 256       // PNT / CAB

typedef float v2f __attribute__((ext_vector_type(2)));
typedef float v8f __attribute__((ext_vector_type(8)));

// ---------------------------------------------------------------------------
// Kernel 1: val[b,r,n] = sum_c x[b,c,n] * w[r,c] + bias[r]
// One wave computes a 16(region) x 16(point) tile with V_WMMA_F32_16X16X4_F32,
// accumulating K = 256 in 64 steps. Block = 256 threads = 8 waves = 128 points.
// Grid = B * (N/128) = 1024 blocks.
//
// A (16x4 f32, M=region, K=channel):  lanes 0-15: M=lane, K={0,1} in vgpr{0,1}
//                                     lanes 16-31: K={2,3}
// B (4x16 f32, K=channel, N=point):   lanes 0-15: N=lane, K={0,1}
//                                     lanes 16-31: N=lane-16, K={2,3}
// D (16x16 f32): vgpr v -> row (v + 8*(lane>=16)), col (lane & 15)
//
// x is loaded with default (RT) temporal hints so it stays resident in the
// 192 MB L2 for the gather pass (x is only 134 MB).
// ---------------------------------------------------------------------------
__global__ void sorter_gemm_wmma(const float* __restrict__ x,
                                 const float* __restrict__ w,
                                 const float* __restrict__ bias,
                                 float* __restrict__ val) {
    const int wid  = threadIdx.x >> 5;
    const int lane = threadIdx.x & 31;
    const int b    = blockIdx.x >> 6;          // 64 tiles of 128 points per batch
    const int nt   = blockIdx.x & 63;
    const int hi   = lane >> 4;                // 0: lanes 0-15, 1: lanes 16-31
    const int m    = lane & 15;                // region row for A / point col for B,D
    const int n    = nt * 128 + wid * 16 + m;  // point index

    const float* xb = x + (size_t)b * CC * NN;

    v8f acc = {};
    for (int k = 0; k < CC; k += 4) {
        const int ka = k + 2 * hi;
        v2f a, bf;
        a.x  = w[m * CC + ka];
        a.y  = w[m * CC + ka + 1];
        bf.x = xb[(size_t)ka * NN + n];
        bf.y = xb[(size_t)(ka + 1) * NN + n];
        acc = __builtin_amdgcn_wmma_f32_16x16x4_f32(
                  /*neg_a=*/false, a, /*neg_b=*/false, bf,
                  /*c_mod=*/(short)0, acc, /*reuse_a=*/false, /*reuse_b=*/false);
    }

#pragma unroll
    for (int v = 0; v < 8; ++v) {
        const int r = v + 8 * hi;
        val[((size_t)b * RR + r) * NN + n] = acc[v] + bias[r];
    }
}

// ---------------------------------------------------------------------------
// Kernel 2: id_activa[b,n] = argmax_r val[b,r,n]  (first max wins, like argmax)
// ---------------------------------------------------------------------------
__global__ void region_argmax(const float* __restrict__ val,
                              float* __restrict__ id_out) {
    const int g = blockIdx.x * 256 + threadIdx.x;   // b*N + n
    const int b = g >> 13;
    const int n = g & (NN - 1);
    const float* base = val + (size_t)b * RR * NN + n;
    float best = base[0];
    int   arg  = 0;
#pragma unroll
    for (int r = 1; r < RR; ++r) {
        const float v = base[(size_t)r * NN];
        if (v > best) { best = v; arg = r; }
    }
    id_out[g] = (float)arg;
}

// ---------------------------------------------------------------------------
// Kernel 3: per-(b,r) descending top-k via full in-LDS bitonic sort.
// key = (monotone_f32_encoding << 32) | ~index  -> descending sort puts the
// largest value first; on value ties the LOWER index wins (matches top_k).
// One block of 512 threads per (b,r) row; 8192 x u64 keys = 64 KB LDS
// (well under the 320 KB/WGP CDNA5 budget).
// ---------------------------------------------------------------------------
__global__ void topk_bitonic(const float* __restrict__ val,
                             int* __restrict__ idx_out) {
    __shared__ unsigned long long keys[NN];
    const int br  = blockIdx.x;          // b*16 + r
    const int tid = threadIdx.x;         // 512 threads
    const float* row = val + (size_t)br * NN;

    for (int i = tid; i < NN; i += 512) {
        const unsigned s   = __float_as_uint(row[i]);
        const unsigned enc = (s & 0x80000000u) ? ~s : (s | 0x80000000u);
        keys[i] = ((unsigned long long)enc << 32) | (unsigned)(~i);
    }
    __syncthreads();

    for (int size = 2; size <= NN; size <<= 1) {
        for (int stride = size >> 1; stride > 0; stride >>= 1) {
            for (int t = tid; t < NN / 2; t += 512) {
                const int i = (t & (stride - 1)) + ((t & ~(stride - 1)) << 1);
                const int j = i + stride;
                const bool desc = ((i & size) == 0);   // overall descending
                const unsigned long long a = keys[i];
                const unsigned long long c = keys[j];
                const bool do_swap = desc ? (a < c) : (a > c);
                if (do_swap) { keys[i] = c; keys[j] = a; }
            }
            __syncthreads();
        }
    }

    for (int p = tid; p < PNT; p += 512) {
        idx_out[(size_t)br * PNT + p] =
            (int)(~(unsigned)(keys[p] & 0xFFFFFFFFu));
    }
}

// ---------------------------------------------------------------------------
// Kernel 4: sp_cube[b,c,r,p] = x[b,c,idx[b,r,p]]  +  cabin segment max.
// One block per (b,r,cabin); threads = 256 points of the cabin.
// - sp_cube (537 MB write-once stream) uses non-temporal stores so it does
//   not evict the L2-resident x (gathers then run at L2 bandwidth).
// - Per-wave maxima are staged in a private LDS row per wave; a SINGLE
//   barrier at the end lets 256 threads fold 8 wave-maxima per channel
//   (vs. 2 barriers per channel previously).
// ---------------------------------------------------------------------------
__global__ void gather_cabins(const float* __restrict__ x,
                              const int* __restrict__ idx,
                              float* __restrict__ sp_cube,
                              float* __restrict__ cabins) {
    __shared__ float wmax[8][CC];        // [wave][channel], 8 KB
    const int blk  = blockIdx.x;
    const int cab  = blk & 7;
    const int br   = blk >> 3;           // b*16 + r
    const int b    = br >> 4;
    const int r    = br & 15;
    const int tid  = threadIdx.x;
    const int lane = tid & 31;
    const int wid  = tid >> 5;
    const int p    = cab * PCAB + tid;

    const int ii = idx[(size_t)br * PNT + p];
    const float* xb = x + (size_t)b * CC * NN;

    for (int c = 0; c < CC; ++c) {
        const float* src = xb + (size_t)c * NN + ii;
        const float v = *src;
        __builtin_prefetch(src + NN, 0, 1);          // global_prefetch_b8
        __builtin_nontemporal_store(
            v, &sp_cube[(((size_t)b * CC + c) * RR + r) * PNT + p]);

        float mx = v;
#pragma unroll
        for (int off = 16; off > 0; off >>= 1)
            mx = fmaxf(mx, __shfl_xor(mx, off, 32));
        if (lane == 0) wmax[wid][c] = mx;            // private row, no race
    }
    __syncthreads();

    // thread t folds the 8 wave-maxima for channel c = t
    {
        const int c = tid;
        float mm = wmax[0][c];
#pragma unroll
        for (int wv = 1; wv < 8; ++wv) mm = fmaxf(mm, wmax[wv][c]);
        cabins[(((size_t)b * CC + c) * RR + r) * CAB + cab] = mm;
    }
}

// ---------------------------------------------------------------------------
// Kernel 5: sp_idx[b,j,r,p] = (float)idx[b,r,p]  for j in [0,19)
// Pure write-once stream -> non-temporal stores.
// ---------------------------------------------------------------------------
__global__ void spidx_bcast(const int* __restrict__ idx,
                            float* __restrict__ out) {
    const int g  = blockIdx.x * 256 + threadIdx.x;   // b*R*pnt
    const int p  = g & (PNT - 1);
    const int br = g >> 11;
    const int b  = br >> 4;
    const int r  = br & 15;
    const float v = (float)idx[g];
#pragma unroll
    for (int j = 0; j < RR + 3; ++j)
        __builtin_nontemporal_store(
            v, &out[(((size_t)b * (RR + 3) + j) * RR + r) * PNT + p]);
}

// ---------------------------------------------------------------------------
extern "C" void kernel_launch(void* const* d_in, const int* in_sizes, int n_in,
                              void* d_out, int out_size, void* d_ws, size_t ws_size,
                              hipStream_t stream) {
    (void)in_sizes; (void)n_in; (void)out_size; (void)ws_size;

    const float* x    = (const float*)d_in[0];   // [16,256,8192]
    const float* w    = (const float*)d_in[1];   // [16,256]
    const float* bias = (const float*)d_in[2];   // [16]

    // Workspace layout: val [B,R,N] f32 (8.39 MB) | idx [B,R,pnt] i32 (2 MB)
    float* ws_val = (float*)d_ws;
    int*   ws_idx = (int*)((char*)d_ws + (size_t)BB * RR * NN * sizeof(float));

    // Output layout (flat f32, reference return order):
    float* out      = (float*)d_out;
    float* sp_cube  = out;                                          // 134,217,728
    float* sp_idx   = sp_cube + (size_t)BB * CC * RR * PNT;         //   9,961,472
    float* cabins   = sp_idx  + (size_t)BB * (RR + 3) * RR * PNT;   //     524,288
    float* id_act   = cabins  + (size_t)BB * CC * RR * CAB;         //     131,072

    // 1. WMMA GEMM + bias -> val
    sorter_gemm_wmma<<<BB * (NN / 128), 256, 0, stream>>>(x, w, bias, ws_val);
    // 2. argmax over regions
    region_argmax<<<(BB * NN) / 256, 256, 0, stream>>>(ws_val, id_act);
    // 3. per-row descending bitonic top-k
    topk_bitonic<<<BB * RR, 512, 0, stream>>>(ws_val, ws_idx);
    // 4. gather sp_cube + cabin segment max
    gather_cabins<<<BB * RR * CAB, 256, 0, stream>>>(x, ws_idx, sp_cube, cabins);
    // 5. sp_idx broadcast
    spidx_bcast<<<(BB * RR * PNT) / 256, 256, 0, stream>>>(ws_idx, sp_idx);
}